// GNNStack_23759759081882
// MI455X (gfx1250) — compile-verified
//
#include <hip/hip_runtime.h>
#include <hip/hip_bf16.h>

typedef __attribute__((ext_vector_type(2))) float v2f;
typedef __attribute__((ext_vector_type(8))) float v8f;

#define DIN 128
#define HID 32
#define BN_EPS 1e-5f

// ---------------------------------------------------------------------------
// deg[c] += w[e]  (in-degree accumulation over real edges; self-loop +1 added
// later in dinv kernel)
// ---------------------------------------------------------------------------
__global__ void gcn_deg_kernel(const long long* __restrict__ ei,
                               const float* __restrict__ ew,
                               float* __restrict__ deg, int E) {
    int e = blockIdx.x * blockDim.x + threadIdx.x;
    if (e >= E) return;
    long long c = ei[(size_t)E + e];
    atomicAdd(&deg[c], ew[e]);
}

// dinv[n] = rsqrt(deg[n] + 1)   (self-loop weight 1; deg >= 1 so always valid)
__global__ void gcn_dinv_kernel(float* __restrict__ deg, int N) {
    int n = blockIdx.x * blockDim.x + threadIdx.x;
    if (n >= N) return;
    deg[n] = rsqrtf(deg[n] + 1.0f);
}

// ---------------------------------------------------------------------------
// Dense transform H[N x 32] = X[N x din] @ W[din x 32] using fp32 WMMA.
// One wave computes one 16-row stripe = two 16x16 accumulators (dout = 32).
// A layout: lane (m = lane&15, kh = lane>>4) holds X[row m][2kh], X[row m][2kh+1]
// B layout: lane holds W[2kh][col m], W[2kh+1][col m] (cols offset per tile)
// C layout: acc[i] = H[row i + 8*kh][col m]
// ---------------------------------------------------------------------------
__global__ void gcn_gemm_wmma(const float* __restrict__ X,
                              const float* __restrict__ W,
                              float* __restrict__ H,
                              int ntiles, int din) {
    int wave = blockIdx.x * (blockDim.x >> 5) + (threadIdx.x >> 5);
    int lane = threadIdx.x & 31;
    if (wave >= ntiles) return;            // wave-uniform: EXEC stays all-ones

    int m  = lane & 15;
    int kh = lane >> 4;

    const float* xrow = X + (size_t)(wave * 16 + m) * din;
    v8f acc0 = {};
    v8f acc1 = {};

    for (int k = 0; k < din; k += 4) {
        int k0 = k + 2 * kh;
        v2f a  = { xrow[k0], xrow[k0 + 1] };
        v2f b0 = { W[k0 * 32 + m],      W[(k0 + 1) * 32 + m] };
        v2f b1 = { W[k0 * 32 + 16 + m], W[(k0 + 1) * 32 + 16 + m] };
        acc0 = __builtin_amdgcn_wmma_f32_16x16x4_f32(
            false, a, false, b0, (short)0, acc0, false, false);
        acc1 = __builtin_amdgcn_wmma_f32_16x16x4_f32(
            false, a, false, b1, (short)0, acc1, false, false);
    }

    float* hbase = H + (size_t)wave * 16 * 32;
#pragma unroll
    for (int i = 0; i < 8; ++i) {
        hbase[(i + 8 * kh) * 32 + m]      = acc0[i];
        hbase[(i + 8 * kh) * 32 + 16 + m] = acc1[i];
    }
}

// ---------------------------------------------------------------------------
// Sparse aggregation: agg[c][j] += dinv[r]*w[e]*dinv[c] * H[r][j]
// One wave per edge; lane j handles feature j. Gather and atomic bursts are
// 128B coalesced; h/agg tables (12.8MB each) are L2-resident.
// ---------------------------------------------------------------------------
__global__ void gcn_aggregate(const long long* __restrict__ ei,
                              const float* __restrict__ ew,
                              const float* __restrict__ dinv,
                              const float* __restrict__ H,
                              float* __restrict__ agg, int E) {
    int t = blockIdx.x * blockDim.x + threadIdx.x;
    int e = t >> 5;
    int j = t & 31;
    if (e >= E) return;
    long long r = ei[e];
    long long c = ei[(size_t)E + e];
    float nrm = dinv[r] * ew[e] * dinv[c];
    atomicAdd(&agg[(size_t)c * 32 + j], nrm * H[(size_t)r * 32 + j]);
}

// ---------------------------------------------------------------------------
// Hidden-layer epilogue pass 1: v = agg + dinv^2 * H (self loop) + bias.
// Written back in-place into agg; per-column sum / sumsq accumulated via
// shared-memory pre-reduction then 64 global atomics per block.
// ---------------------------------------------------------------------------
__global__ void gcn_bias_self_stats(float* __restrict__ agg,
                                    const float* __restrict__ H,
                                    const float* __restrict__ dinv,
                                    const float* __restrict__ bias,
                                    float* __restrict__ stats, int N) {
    __shared__ float ssum[32];
    __shared__ float ssq[32];
    int tid = threadIdx.x;
    if (tid < 32) { ssum[tid] = 0.0f; ssq[tid] = 0.0f; }
    __syncthreads();

    int t = blockIdx.x * blockDim.x + tid;
    int n = t >> 5;
    int j = t & 31;
    if (n < N) {
        float d = dinv[n];
        size_t idx = (size_t)n * 32 + j;
        float v = agg[idx] + d * d * H[idx] + bias[j];
        agg[idx] = v;
        atomicAdd(&ssum[j], v);
        atomicAdd(&ssq[j], v * v);
    }
    __syncthreads();
    if (tid < 32) {
        atomicAdd(&stats[tid],      ssum[tid]);
        atomicAdd(&stats[32 + tid], ssq[tid]);
    }
}

// Hidden-layer epilogue pass 2: batchnorm (biased var) + ReLU -> hcur
__global__ void gcn_bn_relu(const float* __restrict__ v,
                            const float* __restrict__ stats,
                            const float* __restrict__ g,
                            const float* __restrict__ be,
                            float* __restrict__ out, int N) {
    int t = blockIdx.x * blockDim.x + threadIdx.x;
    int n = t >> 5;
    int j = t & 31;
    if (n >= N) return;
    float invN = 1.0f / (float)N;
    float mu   = stats[j] * invN;
    float var  = stats[32 + j] * invN - mu * mu;
    size_t idx = (size_t)n * 32 + j;
    float x = (v[idx] - mu) * rsqrtf(var + BN_EPS) * g[j] + be[j];
    out[idx] = fmaxf(x, 0.0f);
}

// Output-layer epilogue: relu(agg + dinv^2 * H + bias) -> d_out
__global__ void gcn_out_relu(const float* __restrict__ agg,
                             const float* __restrict__ H,
                             const float* __restrict__ dinv,
                             const float* __restrict__ bias,
                             float* __restrict__ out, int N) {
    int t = blockIdx.x * blockDim.x + threadIdx.x;
    int n = t >> 5;
    int j = t & 31;
    if (n >= N) return;
    float d = dinv[n];
    size_t idx = (size_t)n * 32 + j;
    out[idx] = fmaxf(agg[idx] + d * d * H[idx] + bias[j], 0.0f);
}

// ---------------------------------------------------------------------------
extern "C" void kernel_launch(void* const* d_in, const int* in_sizes, int n_in,
                              void* d_out, int out_size, void* d_ws, size_t ws_size,
                              hipStream_t stream) {
    const float*     x  = (const float*)d_in[0];
    const long long* ei = (const long long*)d_in[1];   // int64 edge_index [2,E]
    const float*     ew = (const float*)d_in[2];
    const float* W[4] = { (const float*)d_in[3], (const float*)d_in[5],
                          (const float*)d_in[7], (const float*)d_in[9] };
    const float* B[4] = { (const float*)d_in[4], (const float*)d_in[6],
                          (const float*)d_in[8], (const float*)d_in[10] };
    const float* G[3]  = { (const float*)d_in[11], (const float*)d_in[13],
                           (const float*)d_in[15] };
    const float* BE[3] = { (const float*)d_in[12], (const float*)d_in[14],
                           (const float*)d_in[16] };

    const int N = in_sizes[0] / DIN;    // 100000
    const int E = in_sizes[2];          // 3200000

    // workspace carve-out (floats): dinv | h | agg | hcur | stats(64)
    float* dinv  = (float*)d_ws;
    float* h     = dinv + N;
    float* agg   = h + (size_t)N * 32;
    float* hcur  = agg + (size_t)N * 32;
    float* stats = hcur + (size_t)N * 32;

    const int BLK = 256;
    const int ntiles     = N / 16;                       // 6250, exact
    const int gemm_waves = BLK / 32;
    const int gemm_grid  = (ntiles + gemm_waves - 1) / gemm_waves;
    const int eThreads   = E;                            // one thread / edge
    const int aggThreads = E * 32;                       // wave / edge
    const int nThreads   = N * 32;

    // ---- degree / normalization ----
    hipMemsetAsync(dinv, 0, (size_t)N * sizeof(float), stream);
    gcn_deg_kernel<<<(eThreads + BLK - 1) / BLK, BLK, 0, stream>>>(ei, ew, dinv, E);
    gcn_dinv_kernel<<<(N + BLK - 1) / BLK, BLK, 0, stream>>>(dinv, N);

    for (int l = 0; l < 4; ++l) {
        const float* xin = (l == 0) ? x : hcur;
        const int din    = (l == 0) ? DIN : HID;

        gcn_gemm_wmma<<<gemm_grid, BLK, 0, stream>>>(xin, W[l], h, ntiles, din);

        hipMemsetAsync(agg, 0, (size_t)N * 32 * sizeof(float), stream);
        gcn_aggregate<<<(aggThreads + BLK - 1) / BLK, BLK, 0, stream>>>(
            ei, ew, dinv, h, agg, E);

        if (l < 3) {
            hipMemsetAsync(stats, 0, 64 * sizeof(float), stream);
            gcn_bias_self_stats<<<(nThreads + BLK - 1) / BLK, BLK, 0, stream>>>(
                agg, h, dinv, B[l], stats, N);
            gcn_bn_relu<<<(nThreads + BLK - 1) / BLK, BLK, 0, stream>>>(
                agg, stats, G[l], BE[l], hcur, N);
        } else {
            gcn_out_relu<<<(nThreads + BLK - 1) / BLK, BLK, 0, stream>>>(
                agg, h, dinv, B[l], (float*)d_out, N);
        }
    }
}